// AttnDecoderRNN_50087908606299
// MI455X (gfx1250) — compile-verified
//
#include <hip/hip_runtime.h>
#include <math.h>

typedef __attribute__((ext_vector_type(2))) float v2f;
typedef __attribute__((ext_vector_type(8))) float v8f;

#define H 2048
#define V 50257
#define L 100

// ---------------- K1: embedding gather + attention logits + softmax ----
__global__ void k_attn(const int* __restrict__ tok,
                       const float* __restrict__ hidden,
                       const float* __restrict__ emb,
                       const float* __restrict__ attn_W,
                       const float* __restrict__ attn_b,
                       float* __restrict__ ws_emb,    // ws[0..H)
                       float* __restrict__ attn_out)  // d_out + V + H
{
    __shared__ float xs[2 * H];
    __shared__ float red[256];
    int t = threadIdx.x;
    int row = tok[0];
    for (int k = t; k < H; k += 256) {
        float e = emb[(size_t)row * H + k];
        xs[k] = e;
        ws_emb[k] = e;
        xs[H + k] = hidden[k];
    }
    __syncthreads();

    float logit = -3.0e38f;
    if (t < L) {
        float acc = attn_b[t];
        for (int k = 0; k < 2 * H; ++k)
            acc = fmaf(xs[k], attn_W[k * L + t], acc);
        logit = acc;
    }
    // block max
    red[t] = logit;
    __syncthreads();
    for (int s = 128; s > 0; s >>= 1) {
        if (t < s) red[t] = fmaxf(red[t], red[t + s]);
        __syncthreads();
    }
    float m = red[0];
    __syncthreads();
    float e = (t < L) ? expf(logit - m) : 0.0f;
    red[t] = e;
    __syncthreads();
    for (int s = 128; s > 0; s >>= 1) {
        if (t < s) red[t] += red[t + s];
        __syncthreads();
    }
    float S = red[0];
    if (t < L) attn_out[t] = e / S;
}

// ---------------- K2: context = attn_weights @ encoder_outputs --------
__global__ void k_context(const float* __restrict__ attn_w,  // d_out+V+H
                          const float* __restrict__ enc,     // [L, H]
                          float* __restrict__ ws_ctx)        // ws[H..2H)
{
    __shared__ float w[L];
    int t = threadIdx.x;
    if (t < L) w[t] = attn_w[t];
    __syncthreads();
    int h = blockIdx.x * blockDim.x + t;
    float acc = 0.0f;
#pragma unroll 4
    for (int l = 0; l < L; ++l)
        acc = fmaf(w[l], enc[l * H + h], acc);
    ws_ctx[h] = acc;
}

// ---------------- K3: rnn_input = [emb|ctx] @ comb_W + b  (WMMA f32) ---
// A (16x4) = x chunk broadcast to all 16 rows; B (4x16) = comb_W tile.
// A layout (ISA 7.12.2): lanes 0-15 -> K=0,1 ; lanes 16-31 -> K=2,3.
// D layout: VGPR0 lanes 0-15 = row M=0, N=lane (all rows identical here).
__global__ void k_comb_wmma(const float* __restrict__ x,       // ws[0..2H)
                            const float* __restrict__ comb_W,  // [2H, H]
                            const float* __restrict__ comb_b,
                            float* __restrict__ rnn)           // ws[2H..3H)
{
    __shared__ float xs[2 * H];
    int t = threadIdx.x;
    for (int k = t; k < 2 * H; k += 256) xs[k] = x[k];
    __syncthreads();

    int wave = t >> 5, lane = t & 31;
    int half = lane >> 4, nl = lane & 15;
    int n0 = (blockIdx.x * 8 + wave) * 16;

    v8f c = {};
    for (int k0 = 0; k0 < 2 * H; k0 += 4) {
        int kk = k0 + 2 * half;
        v2f a, b;
        a[0] = xs[kk];
        a[1] = xs[kk + 1];
        const float* bp = comb_W + (size_t)kk * H + n0 + nl;
        b[0] = bp[0];
        b[1] = bp[H];
        c = __builtin_amdgcn_wmma_f32_16x16x4_f32(false, a, false, b,
                                                  (short)0, c, false, false);
    }
    if (lane < 16) {
        int n = n0 + lane;
        rnn[n] = c[0] + comb_b[n];
    }
}

// ---------------- K4: GRU cell, wave-per-output, 6 row dot products ----
__global__ void k_gru(const float* __restrict__ rnn,   // ws+2H
                      const float* __restrict__ h0,    // hidden
                      const float* __restrict__ W_ih, const float* __restrict__ b_ih,
                      const float* __restrict__ W_hh, const float* __restrict__ b_hh,
                      float* __restrict__ h_new)       // d_out + V
{
    int wave = threadIdx.x >> 5, lane = threadIdx.x & 31;
    int j = blockIdx.x * (blockDim.x >> 5) + wave;

    const float4* x4  = (const float4*)rnn;
    const float4* hv4 = (const float4*)h0;
    const float4* wir = (const float4*)(W_ih + (size_t)j * H);
    const float4* wiz = (const float4*)(W_ih + (size_t)(H + j) * H);
    const float4* win = (const float4*)(W_ih + (size_t)(2 * H + j) * H);
    const float4* whr = (const float4*)(W_hh + (size_t)j * H);
    const float4* whz = (const float4*)(W_hh + (size_t)(H + j) * H);
    const float4* whn = (const float4*)(W_hh + (size_t)(2 * H + j) * H);

    float ir = 0, iz = 0, in_ = 0, hr = 0, hz = 0, hn = 0;
    for (int it = lane; it < H / 4; it += 32) {
        float4 xv = x4[it], hv = hv4[it];
        float4 w;
        w = wir[it]; ir  += w.x*xv.x + w.y*xv.y + w.z*xv.z + w.w*xv.w;
        w = wiz[it]; iz  += w.x*xv.x + w.y*xv.y + w.z*xv.z + w.w*xv.w;
        w = win[it]; in_ += w.x*xv.x + w.y*xv.y + w.z*xv.z + w.w*xv.w;
        w = whr[it]; hr  += w.x*hv.x + w.y*hv.y + w.z*hv.z + w.w*hv.w;
        w = whz[it]; hz  += w.x*hv.x + w.y*hv.y + w.z*hv.z + w.w*hv.w;
        w = whn[it]; hn  += w.x*hv.x + w.y*hv.y + w.z*hv.z + w.w*hv.w;
    }
    for (int off = 16; off > 0; off >>= 1) {
        ir  += __shfl_down(ir,  off, 32);
        iz  += __shfl_down(iz,  off, 32);
        in_ += __shfl_down(in_, off, 32);
        hr  += __shfl_down(hr,  off, 32);
        hz  += __shfl_down(hz,  off, 32);
        hn  += __shfl_down(hn,  off, 32);
    }
    if (lane == 0) {
        float r = 1.0f / (1.0f + expf(-(ir + b_ih[j]       + hr + b_hh[j])));
        float z = 1.0f / (1.0f + expf(-(iz + b_ih[H + j]   + hz + b_hh[H + j])));
        float n = tanhf(in_ + b_ih[2 * H + j] + r * (hn + b_hh[2 * H + j]));
        h_new[j] = (1.0f - z) * n + z * h0[j];
    }
}

// ---------------- K5: logits = h_new @ out_W + b ; per-block (m, sumexp)
__global__ void k_logits(const float* __restrict__ hnew,   // d_out+V
                         const float* __restrict__ out_W,  // [H, V]
                         const float* __restrict__ out_b,
                         float* __restrict__ logits,       // ws+3H
                         float* __restrict__ pmax,
                         float* __restrict__ psum)
{
    __shared__ float hs[H];
    __shared__ float red[256];
    int t = threadIdx.x;
    for (int k = t; k < H; k += 256) hs[k] = hnew[k];
    __syncthreads();

    int v = blockIdx.x * 256 + t;
    bool act = v < V;
    int vc = act ? v : V - 1;

    float acc = out_b[vc];
#pragma unroll 8
    for (int h = 0; h < H; ++h) {
        if ((h & 63) == 0 && (h + 64) < H)
            __builtin_prefetch(out_W + (size_t)(h + 64) * V + vc, 0, 0);
        acc = fmaf(hs[h], out_W[(size_t)h * V + vc], acc);
    }
    if (act) logits[v] = acc;

    float logit = act ? acc : -3.0e38f;
    red[t] = logit;
    __syncthreads();
    for (int s = 128; s > 0; s >>= 1) {
        if (t < s) red[t] = fmaxf(red[t], red[t + s]);
        __syncthreads();
    }
    float m = red[0];
    __syncthreads();
    float e = act ? expf(logit - m) : 0.0f;
    red[t] = e;
    __syncthreads();
    for (int s = 128; s > 0; s >>= 1) {
        if (t < s) red[t] += red[t + s];
        __syncthreads();
    }
    if (t == 0) { pmax[blockIdx.x] = m; psum[blockIdx.x] = red[0]; }
}

// ---------------- K6: combine per-block (m, s) pairs -> gmax, logZ -----
__global__ void k_reduce(const float* __restrict__ pmax,
                         const float* __restrict__ psum,
                         int nb, float* __restrict__ red_out)  // [gmax,logZ]
{
    __shared__ float sm[256], ss[256];
    int t = threadIdx.x;
    float m = -3.0e38f, s = 0.0f;
    for (int i = t; i < nb; i += 256) {
        float mi = pmax[i], si = psum[i];
        float mn = fmaxf(m, mi);
        s = s * expf(m - mn) + si * expf(mi - mn);
        m = mn;
    }
    sm[t] = m; ss[t] = s;
    __syncthreads();
    for (int st = 128; st > 0; st >>= 1) {
        if (t < st) {
            float m1 = sm[t], m2 = sm[t + st];
            float mn = fmaxf(m1, m2);
            ss[t] = ss[t] * expf(m1 - mn) + ss[t + st] * expf(m2 - mn);
            sm[t] = mn;
        }
        __syncthreads();
    }
    if (t == 0) { red_out[0] = sm[0]; red_out[1] = logf(ss[0]); }
}

// ---------------- K7: logp = logits - gmax - logZ ----------------------
__global__ void k_final(const float* __restrict__ logits,
                        const float* __restrict__ red,
                        float* __restrict__ logp)
{
    int v = blockIdx.x * 256 + threadIdx.x;
    if (v < V) logp[v] = logits[v] - red[0] - red[1];
}

extern "C" void kernel_launch(void* const* d_in, const int* in_sizes, int n_in,
                              void* d_out, int out_size, void* d_ws, size_t ws_size,
                              hipStream_t stream) {
    const int*   tok    = (const int*)  d_in[0];
    const float* hidden = (const float*)d_in[1];
    const float* enc    = (const float*)d_in[2];
    const float* emb    = (const float*)d_in[3];
    const float* attn_W = (const float*)d_in[4];
    const float* attn_b = (const float*)d_in[5];
    const float* comb_W = (const float*)d_in[6];
    const float* comb_b = (const float*)d_in[7];
    const float* W_ih   = (const float*)d_in[8];
    const float* b_ih   = (const float*)d_in[9];
    const float* W_hh   = (const float*)d_in[10];
    const float* b_hh   = (const float*)d_in[11];
    const float* out_W  = (const float*)d_in[12];
    const float* out_b  = (const float*)d_in[13];

    float* out = (float*)d_out;
    float* ws  = (float*)d_ws;

    float* ws_emb    = ws;                   // H
    float* ws_ctx    = ws + H;               // H  (contiguous with emb)
    float* ws_rnn    = ws + 2 * H;           // H
    float* ws_logits = ws + 3 * H;           // V
    float* ws_pm     = ws + 3 * H + V;       // 256
    float* ws_ps     = ws + 3 * H + V + 256; // 256
    float* ws_red    = ws + 3 * H + V + 512; // 2

    float* out_logp = out;          // V
    float* out_hnew = out + V;      // H
    float* out_attn = out + V + H;  // L

    const int nbV = (V + 255) / 256;  // 197

    k_attn<<<1, 256, 0, stream>>>(tok, hidden, emb, attn_W, attn_b,
                                  ws_emb, out_attn);
    k_context<<<H / 256, 256, 0, stream>>>(out_attn, enc, ws_ctx);
    k_comb_wmma<<<(H / 16) / 8, 256, 0, stream>>>(ws_emb, comb_W, comb_b,
                                                  ws_rnn);
    k_gru<<<H / 8, 256, 0, stream>>>(ws_rnn, hidden, W_ih, b_ih, W_hh, b_hh,
                                     out_hnew);
    k_logits<<<nbV, 256, 0, stream>>>(out_hnew, out_W, out_b,
                                      ws_logits, ws_pm, ws_ps);
    k_reduce<<<1, 256, 0, stream>>>(ws_pm, ws_ps, nbV, ws_red);
    k_final<<<nbV, 256, 0, stream>>>(ws_logits, ws_red, out_logp);
}